// GNN_35304631174084
// MI455X (gfx1250) — compile-verified
//
#include <hip/hip_runtime.h>
#include <hip/hip_bf16.h>
#include <stdint.h>
#include <stddef.h>

typedef __attribute__((ext_vector_type(2))) float v2f;
typedef __attribute__((ext_vector_type(8))) float v8f;

// ---------------------------------------------------------------------------
// Graph-normalization kernels
// ---------------------------------------------------------------------------

__global__ void k_fill1(float* __restrict__ deg, int n) {
    int i = blockIdx.x * 256 + threadIdx.x;
    if (i < n) deg[i] = 1.0f;                       // self-loop weight
}

__global__ void k_deg_accum(const long long* __restrict__ dst,
                            const float* __restrict__ w,
                            float* __restrict__ deg, int E) {
    int e = blockIdx.x * 256 + threadIdx.x;
    if (e < E) atomicAdd(&deg[dst[e]], w[e]);
}

// deg -> dinv (in place), selfnorm = dinv*dinv
__global__ void k_dinv(float* __restrict__ deg, float* __restrict__ selfn, int n) {
    int i = blockIdx.x * 256 + threadIdx.x;
    if (i >= n) return;
    float d = deg[i];
    float di = (d > 0.0f) ? rsqrtf(fmaxf(d, 1e-12f)) : 0.0f;
    deg[i]   = di;
    selfn[i] = di * di;
}

__global__ void k_edge_norm(const long long* __restrict__ src,
                            const long long* __restrict__ dst,
                            const float* __restrict__ w,
                            const float* __restrict__ dinv,
                            float* __restrict__ norm, int E) {
    int e = blockIdx.x * 256 + threadIdx.x;
    if (e < E) norm[e] = dinv[src[e]] * w[e] * dinv[dst[e]];
}

// ---------------------------------------------------------------------------
// Dense GEMM  H = X[nrows,128] @ W[128,128]  via V_WMMA_F32_16X16X4_F32
// 256 threads = 8 waves; block owns 16 rows, wave w owns cols [16w,16w+16)
// ---------------------------------------------------------------------------

__global__ __launch_bounds__(256)
void k_gemm128_wmma(const float* __restrict__ X, const float* __restrict__ W,
                    float* __restrict__ Hout, int nrows) {
    __shared__ float As[16][132];                   // pad 128->132: conflict-free
    int m0 = blockIdx.x << 4;
    int t  = threadIdx.x;

    // cooperative stage of the 16x128 X tile (float4 per thread, 2 iters)
    for (int i = t; i < 512; i += 256) {
        int r = i >> 5;
        int k = (i & 31) << 2;
        float4 v = make_float4(0.f, 0.f, 0.f, 0.f);
        if (m0 + r < nrows)
            v = *(const float4*)(X + (size_t)(m0 + r) * 128 + k);
        *(float4*)&As[r][k] = v;
    }
    __syncthreads();

    int wave = t >> 5;
    int lane = t & 31;
    int nc   = (wave << 4) + (lane & 15);           // output column
    int kb   = (lane >> 4) << 1;                    // K sub-pair: 0 or 2
    int ar   = lane & 15;                           // A row for this lane

    v8f c = {};
    #pragma unroll
    for (int k0 = 0; k0 < 128; k0 += 4) {
        v2f a, b;
        // A 16x4 layout: vgpr0 = K=kb, vgpr1 = K=kb+1
        a.x = As[ar][k0 + kb];
        a.y = As[ar][k0 + kb + 1];
        // B 4x16 layout: row striped across lanes within a VGPR
        b.x = W[(size_t)(k0 + kb)     * 128 + nc];
        b.y = W[(size_t)(k0 + kb + 1) * 128 + nc];
        c = __builtin_amdgcn_wmma_f32_16x16x4_f32(false, a, false, b,
                                                  (short)0, c, false, false);
    }

    // C/D layout: VGPR r holds M = r + 8*(lane>>4), N = lane&15
    int mt = (lane >> 4) << 3;
    #pragma unroll
    for (int r = 0; r < 8; ++r) {
        int m = m0 + mt + r;
        if (m < nrows) Hout[(size_t)m * 128 + nc] = c[r];
    }
}

// ---------------------------------------------------------------------------
// Aggregation kernels (H = 128 path)
// ---------------------------------------------------------------------------

// agg[i,:] = selfnorm[i] * h[i,:]
__global__ void k_self_init(const float* __restrict__ h,
                            const float* __restrict__ selfn,
                            float* __restrict__ agg, int n) {
    int idx = blockIdx.x * 256 + threadIdx.x;       // over n*32 float4s
    if (idx >= n * 32) return;
    int node = idx >> 5;
    int c4   = (idx & 31) << 2;
    float s = selfn[node];
    float4 v = *(const float4*)(h + (size_t)node * 128 + c4);
    v.x *= s; v.y *= s; v.z *= s; v.w *= s;
    *(float4*)(agg + (size_t)node * 128 + c4) = v;
}

// one wave per edge: agg[dst,:] += norm_e * h[src,:]
__global__ __launch_bounds__(256)
void k_edge_scatter(const float* __restrict__ h,
                    const long long* __restrict__ src,
                    const long long* __restrict__ dst,
                    const float* __restrict__ norm,
                    float* __restrict__ agg, int E) {
    int e = blockIdx.x * 8 + (threadIdx.x >> 5);
    if (e >= E) return;
    int lane = threadIdx.x & 31;
    long long s = src[e];
    long long d = dst[e];
    float w = norm[e];
    float4 v = *(const float4*)(h + (size_t)s * 128 + (lane << 2));
    float* o = agg + (size_t)d * 128 + (lane << 2);
    atomicAdd(o + 0, w * v.x);
    atomicAdd(o + 1, w * v.y);
    atomicAdd(o + 2, w * v.z);
    atomicAdd(o + 3, w * v.w);
}

// in-place: x = (relu?) (x + b)
__global__ void k_bias_act(float* __restrict__ x, const float* __restrict__ b,
                           int n, int do_relu) {
    int idx = blockIdx.x * 256 + threadIdx.x;       // over n*32 float4s
    if (idx >= n * 32) return;
    int node = idx >> 5;
    int c4   = (idx & 31) << 2;
    float4 v  = *(float4*)(x + (size_t)node * 128 + c4);
    v.x += b[c4 + 0]; v.y += b[c4 + 1]; v.z += b[c4 + 2]; v.w += b[c4 + 3];
    if (do_relu) {
        v.x = fmaxf(v.x, 0.f); v.y = fmaxf(v.y, 0.f);
        v.z = fmaxf(v.z, 0.f); v.w = fmaxf(v.w, 0.f);
    }
    *(float4*)(x + (size_t)node * 128 + c4) = v;
}

// ---------------------------------------------------------------------------
// Final 128 -> 2 layer
// ---------------------------------------------------------------------------

__global__ void k_gemm2(const float* __restrict__ X, const float* __restrict__ W5,
                        float* __restrict__ Z, int n) {
    int i = blockIdx.x * 256 + threadIdx.x;
    if (i >= n) return;
    const float* row = X + (size_t)i * 128;
    float a0 = 0.f, a1 = 0.f;
    #pragma unroll 8
    for (int k = 0; k < 128; ++k) {
        float v = row[k];
        a0 += v * W5[k * 2 + 0];
        a1 += v * W5[k * 2 + 1];
    }
    Z[(size_t)i * 2 + 0] = a0;
    Z[(size_t)i * 2 + 1] = a1;
}

__global__ void k_self_init2(const float* __restrict__ z,
                             const float* __restrict__ selfn,
                             float* __restrict__ agg2, int n) {
    int i = blockIdx.x * 256 + threadIdx.x;
    if (i >= n) return;
    float s = selfn[i];
    agg2[(size_t)i * 2 + 0] = s * z[(size_t)i * 2 + 0];
    agg2[(size_t)i * 2 + 1] = s * z[(size_t)i * 2 + 1];
}

__global__ void k_edge_scatter2(const float* __restrict__ z,
                                const long long* __restrict__ src,
                                const long long* __restrict__ dst,
                                const float* __restrict__ norm,
                                float* __restrict__ agg2, int E) {
    int e = blockIdx.x * 256 + threadIdx.x;
    if (e >= E) return;
    long long s = src[e];
    long long d = dst[e];
    float w = norm[e];
    atomicAdd(&agg2[(size_t)d * 2 + 0], w * z[(size_t)s * 2 + 0]);
    atomicAdd(&agg2[(size_t)d * 2 + 1], w * z[(size_t)s * 2 + 1]);
}

__global__ void k_bias2(const float* __restrict__ agg2, const float* __restrict__ b5,
                        float* __restrict__ out, int n) {
    int i = blockIdx.x * 256 + threadIdx.x;
    if (i >= n) return;
    out[(size_t)i * 2 + 0] = agg2[(size_t)i * 2 + 0] + b5[0];
    out[(size_t)i * 2 + 1] = agg2[(size_t)i * 2 + 1] + b5[1];
}

// ---------------------------------------------------------------------------
// Launch
// ---------------------------------------------------------------------------

extern "C" void kernel_launch(void* const* d_in, const int* in_sizes, int n_in,
                              void* d_out, int out_size, void* d_ws, size_t ws_size,
                              hipStream_t stream) {
    const float*     x   = (const float*)d_in[0];
    const long long* ei  = (const long long*)d_in[1];   // int64 per reference
    const float*     ea  = (const float*)d_in[2];
    const float*     W1  = (const float*)d_in[3];
    const float*     b1  = (const float*)d_in[4];
    const float*     W2  = (const float*)d_in[5];
    const float*     b2  = (const float*)d_in[6];
    const float*     W3  = (const float*)d_in[7];
    const float*     b3  = (const float*)d_in[8];
    const float*     W4  = (const float*)d_in[9];
    const float*     b4  = (const float*)d_in[10];
    const float*     W5  = (const float*)d_in[11];
    const float*     b5  = (const float*)d_in[12];

    const int N = in_sizes[0] / 128;
    const int E = in_sizes[2];
    const long long* srcp = ei;
    const long long* dstp = ei + E;

    // workspace carve-up (all fp32)
    float* bufA  = (float*)d_ws;                    // [N,128]  layer input / agg
    float* bufB  = bufA  + (size_t)N * 128;         // [N,128]  gemm output
    float* deg   = bufB  + (size_t)N * 128;         // [N]  deg -> dinv
    float* selfn = deg   + N;                       // [N]
    float* normv = selfn + N;                       // [E]
    float* z     = normv + E;                       // [N,2]
    float* agg2  = z     + (size_t)N * 2;           // [N,2]

    const int TB = 256;
    dim3 blkN((N + TB - 1) / TB);
    dim3 blkE((E + TB - 1) / TB);
    dim3 blkNH((N * 32 + TB - 1) / TB);             // N*128/4 float4s
    dim3 blkTiles((N + 15) / 16);
    dim3 blkEdgeWave((E + 7) / 8);                  // 8 edges (waves) per block

    // --- GCN symmetric normalization ---
    k_fill1    <<<blkN, TB, 0, stream>>>(deg, N);
    k_deg_accum<<<blkE, TB, 0, stream>>>(dstp, ea, deg, E);
    k_dinv     <<<blkN, TB, 0, stream>>>(deg, selfn, N);
    k_edge_norm<<<blkE, TB, 0, stream>>>(srcp, dstp, ea, deg, normv, E);

    const float* Ws[4] = {W1, W2, W3, W4};
    const float* bs[4] = {b1, b2, b3, b4};

    // --- hidden layers 1..4 ---
    for (int l = 0; l < 4; ++l) {
        const float* in = (l == 0) ? x : bufA;
        k_gemm128_wmma<<<blkTiles, TB, 0, stream>>>(in, Ws[l], bufB, N);
        k_self_init   <<<blkNH, TB, 0, stream>>>(bufB, selfn, bufA, N);
        k_edge_scatter<<<blkEdgeWave, TB, 0, stream>>>(bufB, srcp, dstp, normv, bufA, E);
        k_bias_act    <<<blkNH, TB, 0, stream>>>(bufA, bs[l], N, (l < 3) ? 1 : 0);
    }

    // --- output layer (128 -> 2) ---
    k_gemm2        <<<blkN, TB, 0, stream>>>(bufA, W5, z, N);
    k_self_init2   <<<blkN, TB, 0, stream>>>(z, selfn, agg2, N);
    k_edge_scatter2<<<blkE, TB, 0, stream>>>(z, srcp, dstp, normv, agg2, E);
    k_bias2        <<<blkN, TB, 0, stream>>>(agg2, b5, (float*)d_out, N);
}